// TwoGCN_30193620091066
// MI455X (gfx1250) — compile-verified
//
#include <hip/hip_runtime.h>
#include <stdint.h>

#define HD 128  // hidden dim == input dim == 128

typedef __attribute__((ext_vector_type(2))) float v2f;
typedef __attribute__((ext_vector_type(8))) float v8f;

// ---------------------------------------------------------------------------
// Utility: zero a float buffer (grid-stride)
// ---------------------------------------------------------------------------
__global__ void zero_f32_kernel(float* __restrict__ p, long long n) {
    long long i = (long long)blockIdx.x * blockDim.x + threadIdx.x;
    long long stride = (long long)gridDim.x * blockDim.x;
    for (; i < n; i += stride) p[i] = 0.0f;
}

// ---------------------------------------------------------------------------
// Degree count: deg[dst[e]] += 1  (deg buffer pre-zeroed)
// ---------------------------------------------------------------------------
__global__ void degree_kernel(const long long* __restrict__ dst,
                              float* __restrict__ deg, int E) {
    int e = blockIdx.x * blockDim.x + threadIdx.x;
    if (e < E) atomicAdd(deg + dst[e], 1.0f);
}

// dinv = (deg + 1)^-1/2   (in place)
__global__ void dinv_kernel(float* __restrict__ deg, int N) {
    int i = blockIdx.x * blockDim.x + threadIdx.x;
    if (i < N) deg[i] = rsqrtf(deg[i] + 1.0f);
}

// ---------------------------------------------------------------------------
// WMMA GEMM: Y[N,128] = X[N,128] @ W[128,128]
// One block = 256 threads = 8 waves; block handles 16 rows, each wave one
// 16x16 output tile (8 waves cover all 128 output columns).
// K=128 accumulated via 32x V_WMMA_F32_16X16X4_F32.
// ---------------------------------------------------------------------------
__global__ __launch_bounds__(256) void gemm128_wmma_kernel(
        const float* __restrict__ X, const float* __restrict__ W,
        float* __restrict__ Y, int N) {
    // 16 rows x 128 cols of X staged in LDS, stride padded to 132 floats so
    // the 16 lanes reading a column hit distinct banks (132 mod 64 = 4).
    __shared__ float ldsx[16 * 132];

    const int tid = threadIdx.x;
    const int rowBase = blockIdx.x * 16;

    // Cooperative tile load: 2048 floats / 256 threads = 8 each.
    for (int i = tid; i < 16 * HD; i += 256) {
        int r = i >> 7, c = i & 127;
        ldsx[r * 132 + c] = X[(long long)(rowBase + r) * HD + c];
    }
    __syncthreads();

    const int wave = tid >> 5;
    const int lane = tid & 31;
    const int half = lane >> 4;   // 0: lanes 0-15, 1: lanes 16-31
    const int l15  = lane & 15;
    const int colBase = wave * 16;

    v8f acc = {};
    #pragma unroll
    for (int k = 0; k < HD; k += 4) {
        const int ka = k + half * 2;   // lanes 0-15: K=k,k+1 ; lanes 16-31: K=k+2,k+3
        v2f a, b;
        a[0] = ldsx[l15 * 132 + ka];
        a[1] = ldsx[l15 * 132 + ka + 1];
        b[0] = W[(long long)ka * HD + colBase + l15];
        b[1] = W[(long long)(ka + 1) * HD + colBase + l15];
        acc = __builtin_amdgcn_wmma_f32_16x16x4_f32(
            /*neg_a=*/false, a, /*neg_b=*/false, b,
            /*c_mod=*/(short)0, acc, /*reuse_a=*/false, /*reuse_b=*/false);
    }

    // C/D layout: VGPR r -> M = r (lanes 0-15) / r+8 (lanes 16-31), N = lane&15
    #pragma unroll
    for (int r = 0; r < 8; ++r) {
        int m = rowBase + r + half * 8;
        Y[(long long)m * HD + colBase + l15] = acc[r];
    }
}

// ---------------------------------------------------------------------------
// Edge scatter: agg[dst] += h[src] * dinv[src]*dinv[dst]
// One wave per edge; each lane moves a float4 (128 channels / 32 lanes).
// ---------------------------------------------------------------------------
__global__ __launch_bounds__(256) void scatter_kernel(
        const float* __restrict__ h, const long long* __restrict__ src,
        const long long* __restrict__ dst, const float* __restrict__ dinv,
        float* __restrict__ agg, int E) {
    int e = blockIdx.x * 8 + (threadIdx.x >> 5);
    if (e >= E) return;
    const int lane = threadIdx.x & 31;
    const long long s = src[e];
    const long long d = dst[e];
    const float w = dinv[s] * dinv[d];
    const float4* hp = (const float4*)h;
    float4 v = hp[s * 32 + lane];
    float* ap = agg + d * HD + lane * 4;
    atomicAdd(ap + 0, v.x * w);
    atomicAdd(ap + 1, v.y * w);
    atomicAdd(ap + 2, v.z * w);
    atomicAdd(ap + 3, v.w * w);
}

// ---------------------------------------------------------------------------
// Fused self-loop + bias + ReLU:  agg = relu(agg + h*dinv^2 + bias)
// One thread per float4.
// ---------------------------------------------------------------------------
__global__ __launch_bounds__(256) void combine_relu_kernel(
        float* __restrict__ agg, const float* __restrict__ h,
        const float* __restrict__ dinv, const float* __restrict__ bias, int N) {
    long long t = (long long)blockIdx.x * 256 + threadIdx.x;  // over N*32 float4s
    if (t >= (long long)N * 32) return;
    int i = (int)(t >> 5);
    int j4 = (int)(t & 31);
    float di = dinv[i];
    float di2 = di * di;
    float4 a  = ((const float4*)agg)[t];
    float4 hv = ((const float4*)h)[t];
    float4 bv = ((const float4*)bias)[j4];
    float4 r;
    r.x = fmaxf(a.x + hv.x * di2 + bv.x, 0.0f);
    r.y = fmaxf(a.y + hv.y * di2 + bv.y, 0.0f);
    r.z = fmaxf(a.z + hv.z * di2 + bv.z, 0.0f);
    r.w = fmaxf(a.w + hv.w * di2 + bv.w, 0.0f);
    ((float4*)agg)[t] = r;
}

// ---------------------------------------------------------------------------
// Final fused: out[i] = dot(agg_i + h_i*dinv^2 + b2, Wo) + bo
// One wave per node, wave32 xor-shuffle reduction.
// ---------------------------------------------------------------------------
__global__ __launch_bounds__(256) void final_kernel(
        const float* __restrict__ agg, const float* __restrict__ h,
        const float* __restrict__ dinv, const float* __restrict__ b2,
        const float* __restrict__ Wo, const float* __restrict__ bo,
        float* __restrict__ out, int N) {
    int i = blockIdx.x * 8 + (threadIdx.x >> 5);
    if (i >= N) return;
    const int lane = threadIdx.x & 31;
    float di = dinv[i];
    float di2 = di * di;
    float4 a  = ((const float4*)agg)[(long long)i * 32 + lane];
    float4 hv = ((const float4*)h)[(long long)i * 32 + lane];
    float4 bb = ((const float4*)b2)[lane];
    float4 w  = ((const float4*)Wo)[lane];
    float p = (a.x + hv.x * di2 + bb.x) * w.x
            + (a.y + hv.y * di2 + bb.y) * w.y
            + (a.z + hv.z * di2 + bb.z) * w.z
            + (a.w + hv.w * di2 + bb.w) * w.w;
    #pragma unroll
    for (int o = 16; o > 0; o >>= 1) p += __shfl_xor(p, o, 32);
    if (lane == 0) out[i] = p + bo[0];
}

// ---------------------------------------------------------------------------
extern "C" void kernel_launch(void* const* d_in, const int* in_sizes, int n_in,
                              void* d_out, int out_size, void* d_ws, size_t ws_size,
                              hipStream_t stream) {
    const float*      x   = (const float*)d_in[0];
    const long long*  ei  = (const long long*)d_in[1];  // int64 (2, E)
    const float*      W1  = (const float*)d_in[2];
    const float*      b1  = (const float*)d_in[3];
    const float*      W2  = (const float*)d_in[4];
    const float*      b2  = (const float*)d_in[5];
    const float*      Wo  = (const float*)d_in[6];
    const float*      bo  = (const float*)d_in[7];
    float*            out = (float*)d_out;

    const int N = in_sizes[0] / HD;     // 100000
    const int E = in_sizes[1] / 2;      // 1600000
    const long long* src = ei;
    const long long* dst = ei + E;

    // Workspace layout: dinv[N] | A[N*HD] | B[N*HD]  (all float, 16B-aligned)
    float* dinv = (float*)d_ws;
    float* A = dinv + N;
    float* B = A + (size_t)N * HD;

    const long long NH = (long long)N * HD;

    // --- degree / normalization ---
    zero_f32_kernel<<<1024, 256, 0, stream>>>(dinv, N);
    degree_kernel<<<(E + 255) / 256, 256, 0, stream>>>(dst, dinv, E);
    dinv_kernel<<<(N + 255) / 256, 256, 0, stream>>>(dinv, N);

    // --- layer 1 ---
    gemm128_wmma_kernel<<<(N + 15) / 16, 256, 0, stream>>>(x, W1, A, N);
    zero_f32_kernel<<<8192, 256, 0, stream>>>(B, NH);
    scatter_kernel<<<(E + 7) / 8, 256, 0, stream>>>(A, src, dst, dinv, B, E);
    combine_relu_kernel<<<(int)((NH / 4 + 255) / 256), 256, 0, stream>>>(B, A, dinv, b1, N);

    // --- layer 2 ---
    gemm128_wmma_kernel<<<(N + 15) / 16, 256, 0, stream>>>(B, W2, A, N);
    zero_f32_kernel<<<8192, 256, 0, stream>>>(B, NH);
    scatter_kernel<<<(E + 7) / 8, 256, 0, stream>>>(A, src, dst, dinv, B, E);

    // --- readout ---
    final_kernel<<<(N + 7) / 8, 256, 0, stream>>>(B, A, dinv, b2, Wo, bo, out, N);
}